// GlobalShiftV2Portion_16930761081413
// MI455X (gfx1250) — compile-verified
//
#include <hip/hip_runtime.h>

// GlobalShiftV2Portion on MI455X (gfx1250).
//
// Reference reduces to:
//   channels [0,256):   out = in                               (identity)
//   channels [256,512): per-channel 128x128 quadrant rotation:
//       q = (ch-256)/64, t_out = (h>=64)*2 + (w>=64), t_in = (q + t_out) & 3
//       out[b,ch,h,w] = in[b,ch, (t_in>>1)*64 + h%64, (t_in&1)*64 + w%64]
//
// Pure permutation: 512 MB traffic -> ~22 us floor at 23.3 TB/s.
// Quadrants are 256B-aligned so every 16-float4 half-row is contiguous in
// both src and dst: perfectly coalesced b128 traffic.
//
// One launch, 8192 blocks x 256 threads:
//   blocks [0,4096):    identity half via CDNA5 async global->LDS->global
//                       (ASYNCcnt-tracked DMA, data never touches VGPRs),
//                       8-deep batched with a descending wait ladder.
//   blocks [4096,8192): shifted half via direct non-temporal b128 gather,
//                       8-way unrolled loads-then-stores.
//
// Index trick: per-thread flat id i0 < 2^20 covers (c,h,w4); the unroll
// index k IS the batch index, adding (k<<21) f4 per step. All address math
// is computed once per thread.

typedef float f4 __attribute__((ext_vector_type(4)));

#define ASYNC_WAIT(N) asm volatile("s_wait_asynccnt " #N ::: "memory")

__global__ __launch_bounds__(256) void global_shift_kernel(
    const f4* __restrict__ x, f4* __restrict__ out) {
  __shared__ __align__(16) f4 lbuf[8 * 256];  // 32 KB: 8 slots x 256 lanes x 16B
  const unsigned tid = threadIdx.x;

  if (blockIdx.x < 4096u) {
    // ---------------- identity half: async DMA path ----------------
    const unsigned i0 = blockIdx.x * 256u + tid;            // < 2^20
    const unsigned la0 = (unsigned)(unsigned long long)(&lbuf[tid]);
    const unsigned long long gs = (unsigned long long)(const void*)x;
    const unsigned long long gd = (unsigned long long)(void*)out;

    unsigned long long ga_s[8], ga_d[8];
    unsigned la[8];
#pragma unroll
    for (unsigned k = 0; k < 8; ++k) {
      const unsigned long long byteoff =
          ((unsigned long long)((k << 21) | i0)) << 4;      // batch k, chans 0..255
      ga_s[k] = gs + byteoff;
      ga_d[k] = gd + byteoff;
      la[k] = la0 + k * 4096u;                              // 16B/lane per slot
    }
    // Issue all 8 async loads back-to-back: 4KB/lane-group in flight.
#pragma unroll
    for (unsigned k = 0; k < 8; ++k)
      asm volatile("global_load_async_to_lds_b128 %0, %1, off"
                   :: "v"(la[k]), "v"(ga_s[k]) : "memory");
    // Descending wait ladder: loads complete in order and at most k stores
    // can have completed before store k, so asynccnt <= 7-k guarantees load k
    // is in LDS while later loads/earlier stores stay in flight.
#define STORE_K(K, W)                                                   \
    ASYNC_WAIT(W);                                                      \
    asm volatile("global_store_async_from_lds_b128 %0, %1, off"         \
                 :: "v"(ga_d[K]), "v"(la[K]) : "memory")
    STORE_K(0, 7); STORE_K(1, 6); STORE_K(2, 5); STORE_K(3, 4);
    STORE_K(4, 3); STORE_K(5, 2); STORE_K(6, 1); STORE_K(7, 0);
#undef STORE_K
    // S_ENDPGM performs an implicit wait-idle; LDS is not reused.
  } else {
    // ---------------- shifted half: direct NT b128 gather ----------------
    const unsigned j0 = (blockIdx.x - 4096u) * 256u + tid;  // < 2^20
    const unsigned w4 = j0 & 31u;           // f4 column (0..31)
    const unsigned h  = (j0 >> 5) & 127u;   // row
    const unsigned lc = (j0 >> 12) & 255u;  // local channel in shifted half
    const unsigned q  = lc >> 6;            // quadrant rotation amount
    const unsigned t_out = ((h >> 6) << 1) | (w4 >> 4);
    const unsigned t_in  = (q + t_out) & 3u;
    const unsigned h_src  = ((t_in >> 1) << 6) | (h & 63u);
    const unsigned w4_src = ((t_in & 1u) << 4) | (w4 & 15u);
    const unsigned half = (1u << 20) | (lc << 12);          // channels 256..511
    const unsigned src_low = half | (h_src << 5) | w4_src;
    const unsigned dst_low = half | (h << 5) | w4;

    f4 v[8];
    // 8 NT b128 loads in flight (streamed once; 256MB > 192MB L2 -> NT).
#pragma unroll
    for (unsigned k = 0; k < 8; ++k)
      v[k] = __builtin_nontemporal_load(x + ((k << 21) | src_low));
#pragma unroll
    for (unsigned k = 0; k < 8; ++k)
      __builtin_nontemporal_store(v[k], out + ((k << 21) | dst_low));
  }
}

extern "C" void kernel_launch(void* const* d_in, const int* in_sizes, int n_in,
                              void* d_out, int out_size, void* d_ws, size_t ws_size,
                              hipStream_t stream) {
  (void)in_sizes; (void)n_in; (void)out_size; (void)d_ws; (void)ws_size;
  const f4* x = (const f4*)d_in[0];
  f4* out = (f4*)d_out;
  // 8192 blocks x 256 threads; each thread moves 8 float4 (128 B).
  // 8192*256*8 = 2^24 float4 = full 256 MB tensor, exact coverage, full EXEC.
  global_shift_kernel<<<8192, 256, 0, stream>>>(x, out);
}